// HeteroVisionConv_29703993819529
// MI455X (gfx1250) — compile-verified
//
#include <hip/hip_runtime.h>
#include <hip/hip_bf16.h>

// ---------------------------------------------------------------------------
// HeteroVisionConv: 2x bipartite GATConv (H=4, D=32, IN=64) + fused out linear
// Strategy: fp32 WMMA (v_wmma_f32_16x16x4_f32) for all dense GEMMs (exact
// precision), 2-pass edge phase with fused unnormalized aggregation, all
// scatter state L2-resident.
// ---------------------------------------------------------------------------

typedef float v2f __attribute__((ext_vector_type(2)));
typedef float v8f __attribute__((ext_vector_type(8)));

static constexpr int NN  = 50000;   // nodes per type (N_AGENT == N_GT == N_UAV)
static constexpr int EE  = 600000;  // edges per relation
static constexpr int KIN = 64;      // input features
static constexpr int HD  = 128;     // H*D = 4*32

// ----- ordered-encoding for float atomic max via unsigned atomicMax --------
__device__ __forceinline__ unsigned fenc(float f) {
  unsigned u = __float_as_uint(f);
  return (u & 0x80000000u) ? ~u : (u | 0x80000000u);
}
__device__ __forceinline__ float fdec(unsigned e) {
  unsigned u = (e & 0x80000000u) ? (e & 0x7FFFFFFFu) : ~e;
  return __uint_as_float(u);
}

// ---------------------------------------------------------------------------
// WMMA fp32 GEMM: C[nrows x 128] = act(A[nrows x K] * B[K x 128] + bias)
// Block = 256 threads = 8 waves; each wave owns a 16(M) x 32(N) tile
// (two 16x16 WMMA accumulators sharing the A fragment).
// EXEC stays all-ones (WMMA requirement): OOB rows are clamped on load and
// masked on store.
// ---------------------------------------------------------------------------
__global__ void __launch_bounds__(256)
gemm_f32_wmma(const float* __restrict__ A, const float* __restrict__ B,
              const float* __restrict__ bias, float* __restrict__ C,
              int nrows, int K, int relu) {
  const int lane  = threadIdx.x & 31;
  const int wave  = threadIdx.x >> 5;
  const int rtile = wave & 1;          // 2 row tiles per block
  const int cpair = wave >> 1;         // 4 column pairs -> 128 cols
  const int row0  = blockIdx.x * 32 + rtile * 16;
  const int col0  = cpair * 32;
  const int m     = lane & 15;
  const int khalf = lane >> 4;         // lanes 0-15: K {0,1}; 16-31: K {2,3}

  int arow = row0 + m;
  if (arow >= nrows) arow = nrows - 1;             // clamp, keep EXEC full
  const float* __restrict__ Arow = A + (size_t)arow * (size_t)K;

  v8f c0 = {};
  v8f c1 = {};
  for (int k = 0; k < K; k += 4) {
    const int kk = k + khalf * 2;
    v2f a;
    a.x = Arow[kk];
    a.y = Arow[kk + 1];
    v2f b0, b1;
    const float* __restrict__ Bk0 = B + (size_t)kk * HD;
    b0.x = Bk0[col0 + m];
    b0.y = Bk0[HD + col0 + m];
    b1.x = Bk0[col0 + 16 + m];
    b1.y = Bk0[HD + col0 + 16 + m];
    // D = A*B + C, fp32 exact  ->  v_wmma_f32_16x16x4_f32
    c0 = __builtin_amdgcn_wmma_f32_16x16x4_f32(false, a, false, b0,
                                               (short)0, c0, false, false);
    c1 = __builtin_amdgcn_wmma_f32_16x16x4_f32(false, a, false, b1,
                                               (short)0, c1, false, false);
  }

  // C/D layout: vgpr r, lanes 0-15 -> M=r, lanes 16-31 -> M=r+8; N = lane&15
  #pragma unroll
  for (int r = 0; r < 8; ++r) {
    const int row = row0 + r + 8 * khalf;
    if (row < nrows) {
      float v0 = c0[r];
      float v1 = c1[r];
      if (bias) { v0 += bias[col0 + m]; v1 += bias[col0 + 16 + m]; }
      if (relu) { v0 = fmaxf(v0, 0.0f); v1 = fmaxf(v1, 0.0f); }
      float* __restrict__ Crow = C + (size_t)row * HD;
      Crow[col0 + m]      = v0;
      Crow[col0 + 16 + m] = v1;
    }
  }
}

// ---------------------------------------------------------------------------
// el/er: out[n][h] = sum_d F[n][h*32+d] * attn[h][d]   (wave per node)
// ---------------------------------------------------------------------------
__global__ void __launch_bounds__(256)
attn_dot(const float* __restrict__ F, const float* __restrict__ attn,
         float* __restrict__ out, int nrows) {
  const int node = blockIdx.x * 8 + (threadIdx.x >> 5);
  if (node >= nrows) return;
  const int lane = threadIdx.x & 31;
  const float4 f = *(const float4*)(F + (size_t)node * HD + lane * 4);
  const float4 a = *(const float4*)(attn + lane * 4);
  float s = f.x * a.x + f.y * a.y + f.z * a.z + f.w * a.w;
  s += __shfl_xor(s, 1, 32);
  s += __shfl_xor(s, 2, 32);
  s += __shfl_xor(s, 4, 32);
  if ((lane & 7) == 0) out[node * 4 + (lane >> 3)] = s;  // 8 lanes per head
}

// ---------------------------------------------------------------------------
// Edge pass 1: e = leaky_relu(el[src]+er[dst]); segment max via atomicMax
// One thread per (edge, head).
// ---------------------------------------------------------------------------
__global__ void __launch_bounds__(256)
edge_pass1(const int* __restrict__ src, const int* __restrict__ dst,
           const float* __restrict__ el, const float* __restrict__ er,
           float* __restrict__ ebuf, unsigned* __restrict__ emax, int nE) {
  const int t = blockIdx.x * 256 + threadIdx.x;
  if (t >= nE * 4) return;
  const int e = t >> 2;
  const int h = t & 3;
  const int s = src[e];
  const int d = dst[e];
  float v = el[s * 4 + h] + er[d * 4 + h];
  v = (v > 0.0f) ? v : 0.2f * v;                 // leaky_relu(., 0.2)
  ebuf[t] = v;
  atomicMax(emax + d * 4 + h, fenc(v));
}

// ---------------------------------------------------------------------------
// Edge pass 2 (fused): ee = exp(e - emax[dst]); denom[dst][h] += ee;
// acc[dst][colOff + c] += ee * fs[src][c]   (unnormalized message sum)
// One wave per edge; lane l: head = l>>3, 4 features per lane.
// ---------------------------------------------------------------------------
__global__ void __launch_bounds__(256)
edge_pass2(const int* __restrict__ src, const int* __restrict__ dst,
           const float* __restrict__ ebuf, const unsigned* __restrict__ emax,
           const float* __restrict__ fs, float* __restrict__ denom,
           float* __restrict__ acc, int colOff, int nE) {
  const int edge = blockIdx.x * 8 + (threadIdx.x >> 5);
  if (edge >= nE) return;
  const int lane = threadIdx.x & 31;
  const int h    = lane >> 3;
  const int s    = src[edge];
  const int d    = dst[edge];
  const float e  = ebuf[edge * 4 + h];
  const float mx = fdec(emax[d * 4 + h]);
  const float ee = __expf(e - mx);
  if ((lane & 7) == 0) atomicAdd(denom + d * 4 + h, ee);
  const int col = h * 32 + (lane & 7) * 4;
  const float4 f = *(const float4*)(fs + (size_t)s * HD + col);
  float* __restrict__ p = acc + (size_t)d * (2 * HD) + colOff + col;
  atomicAdd(p + 0, ee * f.x);
  atomicAdd(p + 1, ee * f.y);
  atomicAdd(p + 2, ee * f.z);
  atomicAdd(p + 3, ee * f.w);
}

// ---------------------------------------------------------------------------
// Finalize both halves of hcat in place:
// hcat[n][c] = relu( (denom>0 ? hcat[n][c]/denom[n][h] : 0) + res[n][c] + b[c] )
// ---------------------------------------------------------------------------
__global__ void __launch_bounds__(256)
finalize_gat(float* __restrict__ hcat,
             const float* __restrict__ denom_gt, const float* __restrict__ denom_uav,
             const float* __restrict__ res_gt,  const float* __restrict__ res_uav,
             const float* __restrict__ b_gt,    const float* __restrict__ b_uav,
             int nrows) {
  const int n = blockIdx.x;
  if (n >= nrows) return;
  const int c    = threadIdx.x;      // 0..255
  const int half = c >> 7;
  const int c128 = c & 127;
  const int h    = c128 >> 5;
  const float* __restrict__ dn  = half ? denom_uav : denom_gt;
  const float* __restrict__ res = half ? res_uav  : res_gt;
  const float* __restrict__ b   = half ? b_uav    : b_gt;
  const float d   = dn[n * 4 + h];
  const float rst = hcat[(size_t)n * 256 + c];
  float v = (d > 0.0f) ? (rst / d) : 0.0f;       // deg-0 dst -> rst = 0
  v += res[(size_t)n * HD + c128] + b[c128];
  hcat[(size_t)n * 256 + c] = fmaxf(v, 0.0f);
}

// ---------------------------------------------------------------------------
extern "C" void kernel_launch(void* const* d_in, const int* in_sizes, int n_in,
                              void* d_out, int out_size, void* d_ws, size_t ws_size,
                              hipStream_t stream) {
  (void)in_sizes; (void)n_in; (void)out_size; (void)ws_size;

  const float* x_gt       = (const float*)d_in[0];
  const float* x_uav      = (const float*)d_in[1];
  const float* x_agent    = (const float*)d_in[2];
  const int*   src_gt     = (const int*)d_in[3];
  const int*   dst_gt     = (const int*)d_in[4];
  const int*   src_uav    = (const int*)d_in[5];
  const int*   dst_uav    = (const int*)d_in[6];
  const float* W_src_gt   = (const float*)d_in[7];
  const float* W_dst_gt   = (const float*)d_in[8];
  const float* attn_l_gt  = (const float*)d_in[9];
  const float* attn_r_gt  = (const float*)d_in[10];
  const float* W_res_gt   = (const float*)d_in[11];
  const float* b_gt       = (const float*)d_in[12];
  const float* W_src_uav  = (const float*)d_in[13];
  const float* W_dst_uav  = (const float*)d_in[14];
  const float* attn_l_uav = (const float*)d_in[15];
  const float* attn_r_uav = (const float*)d_in[16];
  const float* W_res_uav  = (const float*)d_in[17];
  const float* b_uav      = (const float*)d_in[18];
  const float* W_f        = (const float*)d_in[19];
  const float* b_f        = (const float*)d_in[20];
  float*       out        = (float*)d_out;

  // ----- workspace layout (floats) -----------------------------------------
  float* ws = (float*)d_ws;
  size_t o = 0;
  auto alloc = [&](size_t n) { size_t r = o; o += n; return r; };
  // zero-init region (one memset): emax/denom (gt,uav) + concat accumulator
  const size_t o_emax_gt   = alloc(4 * (size_t)NN);
  const size_t o_denom_gt  = alloc(4 * (size_t)NN);
  const size_t o_emax_uav  = alloc(4 * (size_t)NN);
  const size_t o_denom_uav = alloc(4 * (size_t)NN);
  const size_t o_hcat      = alloc(256 * (size_t)NN);
  const size_t zero_floats = o;
  // non-zeroed scratch
  const size_t o_fs_gt   = alloc(128 * (size_t)NN);
  const size_t o_fs_uav  = alloc(128 * (size_t)NN);
  const size_t o_ftmp    = alloc(128 * (size_t)NN);
  const size_t o_res_gt  = alloc(128 * (size_t)NN);
  const size_t o_res_uav = alloc(128 * (size_t)NN);
  const size_t o_el_gt   = alloc(4 * (size_t)NN);
  const size_t o_er_gt   = alloc(4 * (size_t)NN);
  const size_t o_el_uav  = alloc(4 * (size_t)NN);
  const size_t o_er_uav  = alloc(4 * (size_t)NN);
  const size_t o_ebuf    = alloc(4 * (size_t)EE);

  float*    emax_gt_f  = ws + o_emax_gt;   unsigned* emax_gt  = (unsigned*)emax_gt_f;
  float*    denom_gt_p = ws + o_denom_gt;
  float*    emax_uav_f = ws + o_emax_uav;  unsigned* emax_uav = (unsigned*)emax_uav_f;
  float*    denom_uav_p= ws + o_denom_uav;
  float*    hcat       = ws + o_hcat;
  float*    fs_gt      = ws + o_fs_gt;
  float*    fs_uav     = ws + o_fs_uav;
  float*    ftmp       = ws + o_ftmp;
  float*    res_gt     = ws + o_res_gt;
  float*    res_uav    = ws + o_res_uav;
  float*    el_gt      = ws + o_el_gt;
  float*    er_gt      = ws + o_er_gt;
  float*    el_uav     = ws + o_el_uav;
  float*    er_uav     = ws + o_er_uav;
  float*    ebuf       = ws + o_ebuf;

  // ----- grids -------------------------------------------------------------
  const int gGemm  = (NN + 31) / 32;        // 1563
  const int gAttn  = (NN + 7) / 8;          // 6250
  const int gEp1   = (4 * EE + 255) / 256;  // 9375
  const int gEp2   = (EE + 7) / 8;          // 75000
  const dim3 blk(256);

  // 1) zero emax/denom/hcat in one shot
  hipMemsetAsync(ws, 0, zero_floats * sizeof(float), stream);

  // 2) node projections (WMMA fp32 GEMMs) + attention dots
  gemm_f32_wmma<<<gGemm, blk, 0, stream>>>(x_gt,    W_src_gt,  nullptr, fs_gt, NN, KIN, 0);
  gemm_f32_wmma<<<gGemm, blk, 0, stream>>>(x_agent, W_dst_gt,  nullptr, ftmp,  NN, KIN, 0);
  attn_dot<<<gAttn, blk, 0, stream>>>(fs_gt, attn_l_gt, el_gt, NN);
  attn_dot<<<gAttn, blk, 0, stream>>>(ftmp,  attn_r_gt, er_gt, NN);

  gemm_f32_wmma<<<gGemm, blk, 0, stream>>>(x_uav,   W_src_uav, nullptr, fs_uav, NN, KIN, 0);
  gemm_f32_wmma<<<gGemm, blk, 0, stream>>>(x_agent, W_dst_uav, nullptr, ftmp,   NN, KIN, 0);
  attn_dot<<<gAttn, blk, 0, stream>>>(fs_uav, attn_l_uav, el_uav, NN);
  attn_dot<<<gAttn, blk, 0, stream>>>(ftmp,   attn_r_uav, er_uav, NN);

  gemm_f32_wmma<<<gGemm, blk, 0, stream>>>(x_agent, W_res_gt,  nullptr, res_gt,  NN, KIN, 0);
  gemm_f32_wmma<<<gGemm, blk, 0, stream>>>(x_agent, W_res_uav, nullptr, res_uav, NN, KIN, 0);

  // 3) edge phases (gt then uav; ebuf reused)
  edge_pass1<<<gEp1, blk, 0, stream>>>(src_gt, dst_gt, el_gt, er_gt, ebuf, emax_gt, EE);
  edge_pass2<<<gEp2, blk, 0, stream>>>(src_gt, dst_gt, ebuf, emax_gt, fs_gt,
                                       denom_gt_p, hcat, /*colOff=*/0, EE);
  edge_pass1<<<gEp1, blk, 0, stream>>>(src_uav, dst_uav, el_uav, er_uav, ebuf, emax_uav, EE);
  edge_pass2<<<gEp2, blk, 0, stream>>>(src_uav, dst_uav, ebuf, emax_uav, fs_uav,
                                       denom_uav_p, hcat, /*colOff=*/128, EE);

  // 4) per-node finalize: normalize + residual + bias + relu (in place)
  finalize_gat<<<NN, blk, 0, stream>>>(hcat, denom_gt_p, denom_uav_p,
                                       res_gt, res_uav, b_gt, b_uav, NN);

  // 5) output linear: out = relu(hcat @ W_f + b_f), K = 256
  gemm_f32_wmma<<<gGemm, blk, 0, stream>>>(hcat, W_f, b_f, out, NN, 256, 1);
}